// MinibatchDiscrimination1d_72430328481231
// MI455X (gfx1250) — compile-verified
//
#include <hip/hip_runtime.h>
#include <hip/hip_bf16.h>

// Problem constants (from reference)
#define N_ROWS     256      // batch N
#define IN_F       1024     // K
#define OUT_F      64       // B
#define INTER_F    32       // C
#define BC         (OUT_F * INTER_F)   // 2048 = GEMM N-dim
#define OUT_COLS   (IN_F + OUT_F)      // 1088

typedef __attribute__((ext_vector_type(2))) float f32x2;
typedef __attribute__((ext_vector_type(8))) float f32x8;
typedef __attribute__((ext_vector_type(4))) int   i32x4;

// Pointer-to-AS1(int4) and pointer-to-AS3(int4) for the async-LDS builtin.
typedef __attribute__((address_space(1))) i32x4* as1_i32x4_p;
typedef __attribute__((address_space(3))) i32x4* as3_i32x4_p;

// ---------------------------------------------------------------------------
// Kernel 1: M = x @ W   (x: [256,1024], W = T viewed [1024,2048], M: [256,2048])
//
// Block = 256 threads (8 waves). Each block stages a 32-row strip of x into
// LDS via GLOBAL_LOAD_ASYNC_TO_LDS_B128 (ASYNCcnt-tracked), stride-padded to
// 1028 floats so the A-fragment ds_load_b64 pattern hits all 64 banks exactly
// once. Wave w computes column tile tn = blockIdx.y*8 + w for BOTH 16-row
// tiles of the strip, so each B-fragment load feeds two
// V_WMMA_F32_16X16X4_F32 ops (halves W traffic).
//
// A-frag (16x4 f32, ISA 7.12.2): lanes 0-15 row M=lane, VGPR0=K0,VGPR1=K1;
//                                lanes 16-31 row M=lane-16, VGPR0=K2,VGPR1=K3.
// B-frag (4x16 f32, mirrored):   lanes 0-15 col N=lane,  VGPR0=K0,VGPR1=K1;
//                                lanes 16-31 col N=lane-16, VGPR0=K2,VGPR1=K3.
// C/D (16x16 f32): VGPR r -> row (half*8 + r), col = lane&15.
// ---------------------------------------------------------------------------
#define XROWS    32
#define XSTRIDE  1028   // 1024 + 4 floats pad: bank = (row*4 + k) mod 64

__global__ __launch_bounds__(256)
void gemm_wmma_f32_kernel(const float* __restrict__ X,
                          const float* __restrict__ W,
                          float* __restrict__ M)
{
    __shared__ float Xs[XROWS * XSTRIDE];   // ~131.6 KB (<< 320 KB/WGP)

    const int t    = threadIdx.x;
    const int lane = t & 31;
    const int wave = t >> 5;          // 0..7
    const int rg   = blockIdx.x;      // 0..7  row group (32 rows)
    const int cg   = blockIdx.y;      // 0..15 col group (8 tiles of 16)

    // ---- stage X[rg*32 .. rg*32+31, :] into LDS (8 threads per row) ----
    {
        const int row = t >> 3;       // 0..31
        const int sub = t & 7;        // 0..7
        const float* __restrict__ gsrc = X + (size_t)(rg * XROWS + row) * IN_F;
#if __has_builtin(__builtin_amdgcn_global_load_async_to_lds_b128)
        #pragma unroll 4
        for (int j = 0; j < 32; ++j) {
            const int f4 = sub + j * 8;            // float4 index within row
            as1_i32x4_p gp = (as1_i32x4_p)(uintptr_t)(gsrc + f4 * 4);
            as3_i32x4_p lp = (as3_i32x4_p)(uint32_t)(uintptr_t)
                                 &Xs[row * XSTRIDE + f4 * 4];
            __builtin_amdgcn_global_load_async_to_lds_b128(gp, lp, 0, 0);
        }
        asm volatile("s_wait_asynccnt 0x0" ::: "memory");
#else
        #pragma unroll 4
        for (int j = 0; j < 32; ++j) {
            const int f4 = sub + j * 8;
            float4 v = reinterpret_cast<const float4*>(gsrc)[f4];
            *reinterpret_cast<float4*>(&Xs[row * XSTRIDE + f4 * 4]) = v;
        }
#endif
    }
    __syncthreads();

    const int half = lane >> 4;       // 0 or 1
    const int l15  = lane & 15;
    const int koff = half * 2;        // K sub-offset (0 or 2)
    const int tn   = cg * 8 + wave;   // 0..127 column tile
    const int bcol = tn * 16 + l15;

    const float* __restrict__ a0p = &Xs[(l15)      * XSTRIDE + koff];
    const float* __restrict__ a1p = &Xs[(16 + l15) * XSTRIDE + koff];

    f32x8 acc0 = {};
    f32x8 acc1 = {};
    #pragma unroll 4
    for (int k0 = 0; k0 < IN_F; k0 += 4) {
        // B fragment from global (two rows of W, same column)
        f32x2 b;
        b.x = W[(size_t)(k0 + koff + 0) * BC + bcol];
        b.y = W[(size_t)(k0 + koff + 1) * BC + bcol];
        // A fragments from LDS (conflict-free ds_load_b64)
        f32x2 a0; a0.x = a0p[k0]; a0.y = a0p[k0 + 1];
        f32x2 a1; a1.x = a1p[k0]; a1.y = a1p[k0 + 1];
        // 8 args: (neg_a, A, neg_b, B, c_mod, C, reuse_a, reuse_b)
        acc0 = __builtin_amdgcn_wmma_f32_16x16x4_f32(
                   false, a0, false, b, (short)0, acc0, false, false);
        acc1 = __builtin_amdgcn_wmma_f32_16x16x4_f32(
                   false, a1, false, b, (short)0, acc1, false, false);
    }

    // Store both 16x16 D tiles.
    const int r0 = rg * XROWS + half * 8;
    float* __restrict__ d0 = M + (size_t)r0 * BC + tn * 16 + l15;
    float* __restrict__ d1 = d0 + (size_t)16 * BC;
    #pragma unroll
    for (int r = 0; r < 8; ++r) {
        d0[(size_t)r * BC] = acc0[r];
        d1[(size_t)r * BC] = acc1[r];
    }
}

// ---------------------------------------------------------------------------
// Kernel 2: pairwise L1 + exp reduction.
// One block per b (64 blocks), 256 threads (8 waves). Stage M[:, b, :]
// (256 x 32 f32, padded to stride 33) in LDS; j-loop reads are lane-uniform
// broadcasts (conflict-free). out[i, 1024 + b] = sum_j exp(-L1(i,j,b)) - 1
// ---------------------------------------------------------------------------
#define TILE_STRIDE 33
__global__ __launch_bounds__(256)
void pairwise_l1_exp_kernel(const float* __restrict__ M,
                            float* __restrict__ out)
{
    __shared__ float tile[N_ROWS * TILE_STRIDE];

    const int b = blockIdx.x;      // 0..63
    const int i = threadIdx.x;     // 0..255  (one output row per thread)

    const float4* __restrict__ src =
        reinterpret_cast<const float4*>(M + (size_t)i * BC + (size_t)b * INTER_F);
    #pragma unroll
    for (int c4 = 0; c4 < INTER_F / 4; ++c4) {
        float4 v = src[c4];
        tile[i * TILE_STRIDE + c4 * 4 + 0] = v.x;
        tile[i * TILE_STRIDE + c4 * 4 + 1] = v.y;
        tile[i * TILE_STRIDE + c4 * 4 + 2] = v.z;
        tile[i * TILE_STRIDE + c4 * 4 + 3] = v.w;
    }
    __syncthreads();

    float mi[INTER_F];
    #pragma unroll
    for (int c = 0; c < INTER_F; ++c)
        mi[c] = tile[i * TILE_STRIDE + c];

    float acc = 0.0f;
    #pragma unroll 2
    for (int j = 0; j < N_ROWS; ++j) {
        const float* __restrict__ rj = &tile[j * TILE_STRIDE];
        float l1 = 0.0f;
        #pragma unroll
        for (int c = 0; c < INTER_F; ++c)
            l1 += fabsf(mi[c] - rj[c]);
        acc += __expf(-l1);        // j==i contributes exp(0)=1; removed below
    }

    out[(size_t)i * OUT_COLS + IN_F + b] = acc - 1.0f;
}

// ---------------------------------------------------------------------------
// Kernel 3: copy x into out[:, 0:1024] (float4 vectorized).
// ---------------------------------------------------------------------------
__global__ __launch_bounds__(256)
void copy_x_kernel(const float* __restrict__ X, float* __restrict__ out)
{
    const int row = blockIdx.x;    // 0..255
    const int t   = threadIdx.x;   // 0..255, each moves 4 floats
    const float4* __restrict__ src =
        reinterpret_cast<const float4*>(X + (size_t)row * IN_F);
    float4* __restrict__ dst =
        reinterpret_cast<float4*>(out + (size_t)row * OUT_COLS);
    dst[t] = src[t];
}

// ---------------------------------------------------------------------------
extern "C" void kernel_launch(void* const* d_in, const int* in_sizes, int n_in,
                              void* d_out, int out_size, void* d_ws, size_t ws_size,
                              hipStream_t stream)
{
    const float* x = (const float*)d_in[0];   // [256, 1024]
    const float* T = (const float*)d_in[1];   // [1024, 64, 32] == [1024, 2048]
    float* out = (float*)d_out;               // [256, 1088]
    float* M   = (float*)d_ws;                // scratch: [256, 2048] f32 = 2 MB

    // GEMM: 8 row-groups x 16 col-groups, 8 waves per block.
    dim3 ggrid(N_ROWS / XROWS, BC / (8 * 16));
    gemm_wmma_f32_kernel<<<ggrid, 256, 0, stream>>>(x, T, M);

    // Independent of GEMM: copy x into the first 1024 columns.
    copy_x_kernel<<<N_ROWS, 256, 0, stream>>>(x, out);

    // Pairwise reduction consumes M (same stream -> ordered after GEMM).
    pairwise_l1_exp_kernel<<<OUT_F, 256, 0, stream>>>(M, out);
}